// LightGCN_10952166604876
// MI455X (gfx1250) — compile-verified
//
#include <hip/hip_runtime.h>

#define NUM_USERS 100000
#define NUM_ITEMS 50000
#define N_NODES   (NUM_USERS + NUM_ITEMS)
#define LATENT    64
#define NNZ       2400000
#define BATCH     16384

typedef __attribute__((ext_vector_type(2))) float v2f;
typedef __attribute__((ext_vector_type(8))) float v8f;

// ---------------------------------------------------------------------------
// 1) Concatenate [user_emb; item_emb] into x0 (float4 copies, 2.4M quads)
// ---------------------------------------------------------------------------
__global__ void __launch_bounds__(256)
lgcn_concat(const float* __restrict__ ue, const float* __restrict__ ie,
            float* __restrict__ x) {
    int t = blockIdx.x * 256 + threadIdx.x;            // 0 .. N_NODES*16-1
    const int UQ = NUM_USERS * LATENT / 4;             // user quads
    float4* xd = (float4*)x;
    if (t < UQ) {
        xd[t] = ((const float4*)ue)[t];
    } else {
        xd[t] = ((const float4*)ie)[t - UQ];
    }
}

// ---------------------------------------------------------------------------
// 2) Init per-batch accumulators with the layer-0 embeddings
//    t over BATCH*64; b = t>>6, c = t&63
// ---------------------------------------------------------------------------
__global__ void __launch_bounds__(256)
lgcn_init_out(const float* __restrict__ ue, const float* __restrict__ ie,
              const int* __restrict__ uin, const int* __restrict__ iin,
              float* __restrict__ out_u, float* __restrict__ out_i) {
    int t = blockIdx.x * 256 + threadIdx.x;
    int b = t >> 6, c = t & 63;
    out_u[t] = ue[(long long)uin[b] * LATENT + c];
    out_i[t] = ie[(long long)iin[b] * LATENT + c];
}

// ---------------------------------------------------------------------------
// 3) SpMM: y[row] += val * x[col].  One edge per 16 lanes; each lane owns a
//    float4 of the 64-float row.  Table fits in 192MB L2 -> atomics stay in L2.
// ---------------------------------------------------------------------------
__global__ void __launch_bounds__(256)
lgcn_spmm(const int* __restrict__ rows, const int* __restrict__ cols,
          const float* __restrict__ vals, const float* __restrict__ x,
          float* __restrict__ y) {
    long long tid = (long long)blockIdx.x * 256 + threadIdx.x;
    int e = (int)(tid >> 4);           // edge index
    int l = (int)(tid & 15);           // float4 slot (1/16 of the row)
    int r = rows[e];
    int c = cols[e];
    float v = vals[e];
    float4 xv = ((const float4*)(x + (long long)c * LATENT))[l];
    float* yp = y + (long long)r * LATENT + l * 4;
    __hip_atomic_fetch_add(yp + 0, xv.x * v, __ATOMIC_RELAXED, __HIP_MEMORY_SCOPE_AGENT);
    __hip_atomic_fetch_add(yp + 1, xv.y * v, __ATOMIC_RELAXED, __HIP_MEMORY_SCOPE_AGENT);
    __hip_atomic_fetch_add(yp + 2, xv.z * v, __ATOMIC_RELAXED, __HIP_MEMORY_SCOPE_AGENT);
    __hip_atomic_fetch_add(yp + 3, xv.w * v, __ATOMIC_RELAXED, __HIP_MEMORY_SCOPE_AGENT);
}

// ---------------------------------------------------------------------------
// 4) Accumulate this layer's output rows into the per-batch accumulators
// ---------------------------------------------------------------------------
__global__ void __launch_bounds__(256)
lgcn_gather_acc(const float* __restrict__ x,
                const int* __restrict__ uin, const int* __restrict__ iin,
                float* __restrict__ out_u, float* __restrict__ out_i) {
    int t = blockIdx.x * 256 + threadIdx.x;
    int b = t >> 6, c = t & 63;
    out_u[t] += x[(long long)uin[b] * LATENT + c];
    out_i[t] += x[((long long)NUM_USERS + iin[b]) * LATENT + c];
}

// ---------------------------------------------------------------------------
// 5) Batched dot products via chained V_WMMA_F32_16X16X4_F32.
//    Each wave computes D = U_tile(16x64) * I_tile(16x64)^T as 16 k-steps of
//    K=4; preds are diag(D)/16.  A-operand layout (ISA 7.12.2, 16x4 f32):
//    lanes 0-15 hold K={4k,4k+1}, lanes 16-31 hold K={4k+2,4k+3}, M=lane&15.
//    B (4x16) mirrors this with N=lane&15, so both operands are the same
//    float2 fetch pattern from the respective tile.
// ---------------------------------------------------------------------------
__global__ void __launch_bounds__(256)
lgcn_dot_wmma(const float* __restrict__ u, const float* __restrict__ i,
              float* __restrict__ preds) {
    __shared__ float tile[8][16 * 16];
    int lane = threadIdx.x & 31;
    int wave = threadIdx.x >> 5;
    int base = blockIdx.x * 128 + wave * 16;   // first pair of this wave's tile
    int row  = lane & 15;
    int half = lane >> 4;

    const float* up = u + (long long)(base + row) * LATENT + 2 * half;
    const float* ip = i + (long long)(base + row) * LATENT + 2 * half;

    v8f acc = {};
#pragma unroll
    for (int kk = 0; kk < 16; ++kk) {
        v2f a, b;
        a.x = up[kk * 4 + 0];
        a.y = up[kk * 4 + 1];
        b.x = ip[kk * 4 + 0];
        b.y = ip[kk * 4 + 1];
        acc = __builtin_amdgcn_wmma_f32_16x16x4_f32(
            /*neg_a=*/false, a, /*neg_b=*/false, b,
            /*c_mod=*/(short)0, acc, /*reuse_a=*/false, /*reuse_b=*/false);
    }

    // D layout: VGPR r, lanes 0-15 -> (M=r, N=lane); lanes 16-31 -> (M=r+8, N=lane-16)
#pragma unroll
    for (int r = 0; r < 8; ++r) {
        tile[wave][(r + 8 * half) * 16 + row] = acc[r];
    }
    __syncthreads();

    if (lane < 16) {
        // /(N_LAYERS+1) applied to both u and i vectors -> 1/16 on the product
        preds[base + lane] = tile[wave][lane * 16 + lane] * (1.0f / 16.0f);
    }
}

// ---------------------------------------------------------------------------
extern "C" void kernel_launch(void* const* d_in, const int* in_sizes, int n_in,
                              void* d_out, int out_size, void* d_ws, size_t ws_size,
                              hipStream_t stream) {
    const float* ue  = (const float*)d_in[0];
    const float* ie  = (const float*)d_in[1];
    const int*   er  = (const int*)d_in[2];
    const int*   ec  = (const int*)d_in[3];
    const float* ev  = (const float*)d_in[4];
    const int*   uin = (const int*)d_in[5];
    const int*   iin = (const int*)d_in[6];
    float* preds = (float*)d_out;

    const size_t embBytes = (size_t)N_NODES * LATENT * sizeof(float);   // 38.4 MB
    const size_t batBytes = (size_t)BATCH * LATENT * sizeof(float);     // 4 MB
    char* ws = (char*)d_ws;
    float* xbuf  = (float*)(ws);
    float* ybuf  = (float*)(ws + embBytes);
    float* out_u = (float*)(ws + 2 * embBytes);
    float* out_i = (float*)(ws + 2 * embBytes + batBytes);

    // x0 = concat(user_emb, item_emb)
    lgcn_concat<<<(N_NODES * LATENT / 4) / 256, 256, 0, stream>>>(ue, ie, xbuf);
    // acc terms for the batch rows, layer 0
    lgcn_init_out<<<(BATCH * LATENT) / 256, 256, 0, stream>>>(ue, ie, uin, iin, out_u, out_i);

    float* x = xbuf;
    float* y = ybuf;
    for (int layer = 0; layer < 3; ++layer) {
        hipMemsetAsync(y, 0, embBytes, stream);
        lgcn_spmm<<<((long long)NNZ * 16) / 256, 256, 0, stream>>>(er, ec, ev, x, y);
        lgcn_gather_acc<<<(BATCH * LATENT) / 256, 256, 0, stream>>>(y, uin, iin, out_u, out_i);
        float* t = x; x = y; y = t;
    }

    lgcn_dot_wmma<<<BATCH / 128, 256, 0, stream>>>(out_u, out_i, preds);
}